// Permutation_81037442941340
// MI455X (gfx1250) — compile-verified
//
#include <hip/hip_runtime.h>
#include <math.h>

// x: (8192, 4096) f32, logit: (11,3) f32, out: (8192, 4096) f32
#define N      4096
#define NSTAGE 11
#define TPB    64
#define EPT    64                                  // TPB*EPT == N
#define PADW(k) ((k) + ((((k) >> 5)) << 2))        // +4 words per 32-word block (keeps 16B align)
#define NPAD   (N + ((N >> 5) << 2))               // 4608 floats per buffer

__global__ __launch_bounds__(TPB)
void perm_mult_kernel(const float* __restrict__ logit,
                      const float* __restrict__ x,
                      float* __restrict__ out)
{
    __shared__ __align__(16) float bufA[NPAD];
    __shared__ __align__(16) float bufB[NPAD];
    __shared__ float scoef[NSTAGE][2][4];   // c0..c3 per (stage, half)

    const int t   = threadIdx.x;
    const int row = blockIdx.x;

    // ---- per-stage blend coefficients: z = c0*xa + c1*xb + c2*xar + c3*xbr ----
    if (t < NSTAGE * 2) {
        const int ls = t >> 1, h = t & 1;
        const float p0 = 1.0f / (1.0f + __expf(-logit[ls * 3]));
        const float ph = 1.0f / (1.0f + __expf(-logit[ls * 3 + 1 + h]));
        const float cid = 1.0f - p0, w = 1.0f - ph;
        scoef[ls][h][0] = w * cid;
        scoef[ls][h][1] = w * p0;
        scoef[ls][h][2] = ph * cid;
        scoef[ls][h][3] = ph * p0;
    }

    // ---- async global -> LDS copy of the row into swizzled bufA (ASYNCcnt path) ----
    unsigned ldsA = (unsigned)(size_t)(void*)bufA;     // low 32b == LDS offset
    unsigned long long srow = (unsigned long long)(size_t)(x + (size_t)row * N);
    {
        const int k0 = t * EPT;
#pragma unroll
        for (int c = 0; c < EPT / 4; ++c) {
            const int k   = k0 + 4 * c;                // 4-float granule, stays in one 32-block
            unsigned ldsd = ldsA + 4u * (unsigned)PADW(k);
            unsigned voff = 4u * (unsigned)k;
            asm volatile("global_load_async_to_lds_b128 %0, %1, %2"
                         :: "v"(ldsd), "v"(voff), "s"(srow)
                         : "memory");
        }
    }
    asm volatile("s_wait_asynccnt 0" ::: "memory");
    __syncthreads();

    // ---- one LDS butterfly stage, ls in [6,10] ----
    // Thread owns (group g, BOTH halves h, j-block of 16 front pairs + their mirrors) = 64 outputs.
    // Mirror pairs share all inputs -> 2 LDS reads/elem; all runs contiguous -> b128 merged.
    auto stage = [&](const float* __restrict__ S, float* __restrict__ D, int ls) {
        const int s    = 1 << ls;
        const int g    = t >> (ls - 5);
        const int m    = t & ((s >> 5) - 1);
        const int j0   = m << 4;
        const int base = g << (ls + 1);

        // even/odd windows: 32 consecutive words each (cover both halves' B terms)
        float xb[32], xbr[32];
        {
            const float4* pb  = (const float4*)(S + PADW(base + 2 * j0));
            const float4* pbr = (const float4*)(S + PADW(base + 2 * (s - 16 - j0)));
#pragma unroll
            for (int c = 0; c < 8; ++c) {
                *(float4*)&xb[4 * c]  = pb[c];
                *(float4*)&xbr[4 * c] = pbr[c];
            }
        }

#pragma unroll
        for (int h = 0; h < 2; ++h) {
            const float c0 = scoef[ls][h][0], c1 = scoef[ls][h][1];
            const float c2 = scoef[ls][h][2], c3 = scoef[ls][h][3];

            float xa[16], xr[16];
            {
                const float4* pa = (const float4*)(S + PADW(base + h * s + j0));
                const float4* pr = (const float4*)(S + PADW(base + h * s + (s - 16 - j0)));
#pragma unroll
                for (int c = 0; c < 4; ++c) {
                    *(float4*)&xa[4 * c] = pa[c];
                    *(float4*)&xr[4 * c] = pr[c];
                }
            }

            float zf[16], zm[16];
#pragma unroll
            for (int d = 0; d < 16; ++d) {
                const float A  = xa[d];
                const float AR = xr[15 - d];
                const float B  = xb[2 * d + h];
                const float BR = xbr[30 - 2 * d + h];
                zf[d]      = c0 * A  + c1 * B  + c2 * AR + c3 * BR;  // out at j
                zm[15 - d] = c0 * AR + c1 * BR + c2 * A  + c3 * B;   // out at s-1-j
            }

            float4* df = (float4*)(D + PADW(base + h * s + j0));
            float4* dm = (float4*)(D + PADW(base + h * s + (s - 16 - j0)));
#pragma unroll
            for (int c = 0; c < 4; ++c) {
                df[c] = *(const float4*)&zf[4 * c];
                dm[c] = *(const float4*)&zm[4 * c];
            }
        }
    };

    // stages ls = 10..6 via LDS ping-pong
    stage(bufA, bufB, 10); __syncthreads();
    stage(bufB, bufA, 9);  __syncthreads();
    stage(bufA, bufB, 8);  __syncthreads();
    stage(bufB, bufA, 7);  __syncthreads();
    stage(bufA, bufB, 6);  __syncthreads();   // ownership differs from chunks -> barrier

    // ---- pull contiguous 64-chunk into registers ----
    float v[EPT];
    {
        const float4* p0 = (const float4*)(bufB + PADW(EPT * t));
        const float4* p1 = (const float4*)(bufB + PADW(EPT * t + 32));
#pragma unroll
        for (int c = 0; c < 8; ++c) {
            *(float4*)&v[4 * c]      = p0[c];
            *(float4*)&v[32 + 4 * c] = p1[c];
        }
    }

    // ---- stages ls = 5..0 entirely in registers (group <= 64 == chunk) ----
#pragma unroll
    for (int ls = 5; ls >= 0; --ls) {
        const int ss = 1 << ls;
        const float a0 = scoef[ls][0][0], a1 = scoef[ls][0][1];
        const float a2 = scoef[ls][0][2], a3 = scoef[ls][0][3];
        const float b0 = scoef[ls][1][0], b1 = scoef[ls][1][1];
        const float b2 = scoef[ls][1][2], b3 = scoef[ls][1][3];
        float nv[EPT];
#pragma unroll
        for (int g0 = 0; g0 < EPT; g0 += 2 * ss) {
#pragma unroll
            for (int h = 0; h < 2; ++h) {
                const float c0 = h ? b0 : a0, c1 = h ? b1 : a1;
                const float c2 = h ? b2 : a2, c3 = h ? b3 : a3;
#pragma unroll
                for (int j = 0; j < ss; ++j) {
                    const int jr = ss - 1 - j;
                    nv[g0 + h * ss + j] =
                        c0 * v[g0 + h * ss + j] + c1 * v[g0 + 2 * j + h] +
                        c2 * v[g0 + h * ss + jr] + c3 * v[g0 + 2 * jr + h];
                }
            }
        }
#pragma unroll
        for (int i = 0; i < EPT; ++i) v[i] = nv[i];
    }

    // ---- vectorized global store ----
    float4* po = (float4*)(out + (size_t)row * N + (size_t)t * EPT);
#pragma unroll
    for (int c = 0; c < EPT / 4; ++c) {
        po[c] = make_float4(v[4 * c], v[4 * c + 1], v[4 * c + 2], v[4 * c + 3]);
    }
}

extern "C" void kernel_launch(void* const* d_in, const int* in_sizes, int n_in,
                              void* d_out, int out_size, void* d_ws, size_t ws_size,
                              hipStream_t stream) {
    const float* logit = (const float*)d_in[0];   // (11,3) f32
    const float* x     = (const float*)d_in[1];   // (8192,4096) f32
    float* out         = (float*)d_out;

    const int batch = in_sizes[1] / N;            // 8192 rows
    perm_mult_kernel<<<dim3(batch), dim3(TPB), 0, stream>>>(logit, x, out);
}